// RGCN_17119739641938
// MI455X (gfx1250) — compile-verified
//
#include <hip/hip_runtime.h>
#include <math.h>

typedef float v2f __attribute__((ext_vector_type(2)));
typedef float v8f __attribute__((ext_vector_type(8)));

#if defined(__HIP_DEVICE_COMPILE__) && \
    __has_builtin(__builtin_amdgcn_global_load_async_to_lds_b128) && \
    __has_builtin(__builtin_amdgcn_s_wait_asynccnt)
#define USE_ASYNC_COPY 1
// Builtin signature (from hipcc diagnostic): arg0 = int4 addrspace(1)*, then LDS ptr, imm, imm.
typedef int b128_t __attribute__((vector_size(4 * sizeof(int))));
typedef __attribute__((address_space(1))) b128_t gb128;
typedef __attribute__((address_space(3))) b128_t lb128;
#else
#define USE_ASYNC_COPY 0
#endif

// ---------------------------------------------------------------------------
// One-shot weight pre-swizzle into (k-pair, n) interleaved order:
//   Wsw[by][ (k>>1)*128 + n*2 + (k&1) ] = W_by[k][n]
// so (B[k][n], B[k+1][n]) is a contiguous, 8B-aligned float2 -> the GEMM inner
// loop gets ds_load_2addr_b64 directly into even VGPR pairs (no v_mov fixup),
// and chunks stay contiguous for linear b128 async DMA.
// ---------------------------------------------------------------------------
template <int DIN>
__global__ __launch_bounds__(256) void swizzle_w_kernel(
    const float* __restrict__ Wrel,   // [4][DIN][64]
    const float* __restrict__ Wself,  // [DIN][64]
    float* __restrict__ Wsw)          // [5][DIN*64] swizzled
{
    int idx = blockIdx.x * 256 + threadIdx.x;
    if (idx >= 5 * DIN * 64) return;
    int by  = idx / (DIN * 64);
    int rem = idx - by * DIN * 64;
    int k = rem >> 6, n = rem & 63;
    const float* Sp = (by < 4) ? (Wrel + (size_t)by * DIN * 64) : Wself;
    Wsw[(size_t)by * DIN * 64 + (k >> 1) * 128 + (n << 1) + (k & 1)] = Sp[rem];
}

// ---------------------------------------------------------------------------
// Batched GEMM: for weight block by = blockIdx.y:
//   by<4 : Ycat[:, by*64:(by+1)*64] = A @ Wrel[by]     (ldc=256)
//   by=4 : ACC                      = A @ Wself        (ldc=64, initializes ACC)
// A: [M][DIN] f32.  128 threads = 4 wave32, 64x64 tile, V_WMMA_F32_16X16X4_F32.
// A and swizzled-B chunks staged via GLOBAL_LOAD_ASYNC_TO_LDS_B128 (ASYNCcnt).
// ---------------------------------------------------------------------------
template <int DIN>
__global__ __launch_bounds__(128) void gemm_cat_kernel(
    const float* __restrict__ A,      // [M][DIN]
    const float* __restrict__ Wsw,    // [5][DIN*64] pre-swizzled weights
    float* __restrict__ Ycat,         // [M][256]
    float* __restrict__ ACC,          // [M][64]
    int M)
{
    constexpr int KC  = 64;           // K chunk staged in LDS
    constexpr int LDA = KC + 4;       // pad: row advance = 4 banks -> conflict-free
    __shared__ float Alds[64 * LDA];  // 17408 B
    __shared__ float Wlds[KC * 64];   // 16384 B, swizzled chunk

    const int tid = threadIdx.x;
    const int m0  = blockIdx.x * 64;
    const int by  = blockIdx.y;
    const float* Bp = Wsw + (size_t)by * DIN * 64;

    const int wave = tid >> 5;
    const int lane = tid & 31;
    const int l16  = lane & 15;
    const int lh   = lane >> 4;       // half-wave select

    v8f acc[4] = {};

    for (int kb = 0; kb < DIN; kb += KC) {
        if (kb) __syncthreads();
#if USE_ASYNC_COPY
        // Direct memory->LDS DMA, no VGPR round trip; one asynccnt wait.
        #pragma unroll
        for (int i = 0; i < 8; ++i) {
            int idx = i * 128 + tid;              // 0..1023 b128 chunks
            int m  = idx >> 4;
            int kc = (idx & 15) << 2;
            int gm = m0 + m; gm = (gm < M) ? gm : (M - 1);
            __builtin_amdgcn_global_load_async_to_lds_b128(
                (gb128*)(A + (long long)gm * DIN + kb + kc),
                (lb128*)&Alds[m * LDA + kc], 0, 0);
        }
        #pragma unroll
        for (int i = 0; i < 8; ++i) {
            int idx = i * 128 + tid;              // linear copy of swizzled chunk
            __builtin_amdgcn_global_load_async_to_lds_b128(
                (gb128*)(Bp + (size_t)kb * 64 + (idx << 2)),
                (lb128*)&Wlds[idx << 2], 0, 0);
        }
        __builtin_amdgcn_s_wait_asynccnt(0);
#else
        // Fallback: two-phase (batch loads, then batch LDS stores) to avoid
        // per-element load->wait->store serialization.
        float4 ta[8], tw[8];
        #pragma unroll
        for (int i = 0; i < 8; ++i) {
            int idx = i * 128 + tid;
            int m  = idx >> 4;
            int kc = (idx & 15) << 2;
            int gm = m0 + m; gm = (gm < M) ? gm : (M - 1);
            ta[i] = *(const float4*)(A + (long long)gm * DIN + kb + kc);
            tw[i] = *(const float4*)(Bp + (size_t)kb * 64 + (idx << 2));
        }
        #pragma unroll
        for (int i = 0; i < 8; ++i) {
            int idx = i * 128 + tid;
            int m  = idx >> 4;
            int kc = (idx & 15) << 2;
            *(float4*)&Alds[m * LDA + kc] = ta[i];
            *(float4*)&Wlds[idx << 2] = tw[i];
        }
#endif
        __syncthreads();

        // A 16x4: lanes 0-15 rows, K={0,1}; lanes 16-31 K={2,3}.
        // B 4x16: VGPR0 = K {0|2}, VGPR1 = K {1|3}; swizzled layout makes each
        // (b.x, b.y) a contiguous aligned float2 -> ds_load_2addr_b64.
        const float* Aw = &Alds[(wave * 16 + l16) * LDA + (lh << 1)];
        const float* Bw = &Wlds[(lh << 7) + (l16 << 1)];   // (2*lh)*64 + l16*2
        #pragma unroll
        for (int ks = 0; ks < KC / 4; ++ks) {
            v2f a = *(const v2f*)(Aw + (ks << 2));
            const float* Bk = Bw + (ks << 8);              // advance 4 K rows
            v2f b0 = *(const v2f*)(Bk +  0);
            v2f b1 = *(const v2f*)(Bk + 32);
            v2f b2 = *(const v2f*)(Bk + 64);
            v2f b3 = *(const v2f*)(Bk + 96);
            acc[0] = __builtin_amdgcn_wmma_f32_16x16x4_f32(false, a, false, b0, (short)0, acc[0], false, false);
            acc[1] = __builtin_amdgcn_wmma_f32_16x16x4_f32(false, a, false, b1, (short)0, acc[1], false, false);
            acc[2] = __builtin_amdgcn_wmma_f32_16x16x4_f32(false, a, false, b2, (short)0, acc[2], false, false);
            acc[3] = __builtin_amdgcn_wmma_f32_16x16x4_f32(false, a, false, b3, (short)0, acc[3], false, false);
        }
    }

    // Epilogue: plain store (no C read anywhere in the pipeline).
    float* Cp; int ldc;
    if (by < 4) { Cp = Ycat + by * 64; ldc = 256; }
    else        { Cp = ACC;            ldc = 64;  }
    const int rbase = m0 + wave * 16 + (lh << 3);
    #pragma unroll
    for (int nt = 0; nt < 4; ++nt) {
        int n = nt * 16 + l16;
        #pragma unroll
        for (int j = 0; j < 8; ++j) {
            int gm = rbase + j;
            if (gm < M) Cp[(long long)gm * ldc + n] = acc[nt][j];
        }
    }
}

// ---------------------------------------------------------------------------
// Fused 4-relation scatter in output space:
//   ACC[dst[r][e], :] += sign_r * Ycat[src[r][e], r*64 : r*64+64]
// 64-wide atomics (half the layer-1 traffic of scattering raw embeddings).
// ---------------------------------------------------------------------------
__global__ __launch_bounds__(256) void scatter_y_kernel(
    const float* __restrict__ Ycat, const int* __restrict__ src,
    const int* __restrict__ dst, float* __restrict__ ACC, int E)
{
    const int r = blockIdx.y;
    long long t = (long long)blockIdx.x * blockDim.x + threadIdx.x;
    if (t >= (long long)E * 16) return;
    int e  = (int)(t >> 4);
    int c4 = ((int)t & 15) << 2;
    long long sr = src[(size_t)r * E + e];
    long long dr = dst[(size_t)r * E + e];
    const float s = (r == 3) ? -1.f : 1.f;   // SUBTRACT_REL == 3
    float4 v = *(const float4*)(Ycat + sr * 256 + (r << 6) + c4);
    float* p = ACC + dr * 64 + c4;
    atomicAdd(p + 0, s * v.x);
    atomicAdd(p + 1, s * v.y);
    atomicAdd(p + 2, s * v.z);
    atomicAdd(p + 3, s * v.w);
}

// ---------------------------------------------------------------------------
// Elementwise activation: mode 0 = ReLU, mode 1 = sigmoid (float4).
// ---------------------------------------------------------------------------
__global__ __launch_bounds__(256) void act_kernel(
    const float* __restrict__ in, float* __restrict__ out, long long n4, int mode)
{
    long long i = (long long)blockIdx.x * blockDim.x + threadIdx.x;
    if (i >= n4) return;
    float4 v = ((const float4*)in)[i];
    if (mode == 0) {
        v.x = fmaxf(v.x, 0.f); v.y = fmaxf(v.y, 0.f);
        v.z = fmaxf(v.z, 0.f); v.w = fmaxf(v.w, 0.f);
    } else {
        v.x = 1.f / (1.f + __expf(-v.x)); v.y = 1.f / (1.f + __expf(-v.y));
        v.z = 1.f / (1.f + __expf(-v.z)); v.w = 1.f / (1.f + __expf(-v.w));
    }
    ((float4*)out)[i] = v;
}

// ---------------------------------------------------------------------------
// Per layer (3 kernels): batched GEMM (Ycat + ACC init) -> fused scatter -> act.
// Uses   segment_sum(h[src]) @ W == segment_sum((h @ W)[src])   to scatter in
// the narrow 64-dim space. Whole working set (~180 MB) is L2-resident on MI455X.
// ---------------------------------------------------------------------------
extern "C" void kernel_launch(void* const* d_in, const int* in_sizes, int n_in,
                              void* d_out, int out_size, void* d_ws, size_t ws_size,
                              hipStream_t stream) {
    (void)in_sizes; (void)n_in; (void)out_size; (void)ws_size;
    const float* node_emb = (const float*)d_in[0];
    const int*   src      = (const int*)d_in[1];
    const int*   dst      = (const int*)d_in[2];
    const float* W1       = (const float*)d_in[3];
    const float* Wself1   = (const float*)d_in[4];
    const float* W2       = (const float*)d_in[5];
    const float* Wself2   = (const float*)d_in[6];
    float* out = (float*)d_out;

    const int N = 100000, E = 150000;
    float* Ycat = (float*)d_ws;                 // N*256 floats
    float* ACC  = Ycat + (size_t)N * 256;       // N*64
    float* H1   = ACC  + (size_t)N * 64;        // N*64
    float* Wsw1 = H1   + (size_t)N * 64;        // 5*128*64
    float* Wsw2 = Wsw1 + (size_t)5 * 128 * 64;  // 5*64*64

    dim3 gGemm((N + 63) / 64, 5);
    dim3 gScat((int)(((long long)E * 16 + 255) / 256), 4);
    int  gAct = (int)(((long long)N * 16 + 255) / 256);

    // One-shot weight swizzle (240 KB total).
    swizzle_w_kernel<128><<<(5 * 128 * 64 + 255) / 256, 256, 0, stream>>>(W1, Wself1, Wsw1);
    swizzle_w_kernel<64><<<(5 * 64 * 64 + 255) / 256, 256, 0, stream>>>(W2, Wself2, Wsw2);

    // ---------------- layer 1 (din = 128) ----------------
    gemm_cat_kernel<128><<<gGemm, 128, 0, stream>>>(node_emb, Wsw1, Ycat, ACC, N);
    scatter_y_kernel<<<gScat, 256, 0, stream>>>(Ycat, src, dst, ACC, E);
    act_kernel<<<gAct, 256, 0, stream>>>(ACC, H1, (long long)N * 16, 0);

    // ---------------- layer 2 (din = 64) ----------------
    gemm_cat_kernel<64><<<gGemm, 128, 0, stream>>>(H1, Wsw2, Ycat, ACC, N);
    scatter_y_kernel<<<gScat, 256, 0, stream>>>(Ycat, src, dst, ACC, E);
    act_kernel<<<gAct, 256, 0, stream>>>(ACC, out, (long long)N * 16, 1);
}